// SphericalPoseNet_53188874994220
// MI455X (gfx1250) — compile-verified
//
#include <hip/hip_runtime.h>
#include <math.h>

// ---------------- CDNA5 WMMA types / wrapper ----------------
typedef float v2f __attribute__((ext_vector_type(2)));
typedef float v8f __attribute__((ext_vector_type(8)));

__device__ __forceinline__ v8f wmma4_f32(v2f a, v2f b, v8f c) {
    // V_WMMA_F32_16X16X4_F32 : D = A(16x4) * B(4x16) + C, full f32
    return __builtin_amdgcn_wmma_f32_16x16x4_f32(
        /*neg_a=*/false, a, /*neg_b=*/false, b,
        /*c_mod=*/(short)0, c, /*reuse_a=*/false, /*reuse_b=*/false);
}

#define N_PTS   8192
#define C_DIM   256
#define KC      128
#define KF      16
#define STRIP   1024
#define TEMP_INV (1.0f / 0.07f)
#define LOCAL_RAD_F 0.2617993877991494f  // radians(15)

// ---------------- 1) projections: q = A@Wq + bq, k = B@Wk + bk ----------------
// grid (C/16, N/16, 2), block 32 (one wave -> one 16x16 output tile)
__global__ void proj_kernel(const float* __restrict__ fa, const float* __restrict__ fb,
                            const float* __restrict__ Wq, const float* __restrict__ bq,
                            const float* __restrict__ Wk, const float* __restrict__ bk,
                            float* __restrict__ qo, float* __restrict__ ko) {
    const float* in;  const float* W;  const float* bias;  float* out;
    if (blockIdx.z == 0) { in = fa; W = Wq; bias = bq; out = qo; }
    else                 { in = fb; W = Wk; bias = bk; out = ko; }

    const int lane = threadIdx.x;
    const int half = lane >> 4;
    const int l    = lane & 15;
    const int rowbase = blockIdx.y * 16;
    const int colbase = blockIdx.x * 16;

    v8f acc = {};
    const float* arow = in + (size_t)(rowbase + l) * C_DIM;
    #pragma unroll 4
    for (int kk = 0; kk < C_DIM; kk += 4) {
        const int ka = kk + 2 * half;            // lanes 0-15: K=kk,kk+1 ; 16-31: kk+2,kk+3
        v2f a = *(const v2f*)(arow + ka);
        v2f b;
        b.x = W[(size_t)ka       * C_DIM + colbase + l];
        b.y = W[(size_t)(ka + 1) * C_DIM + colbase + l];
        acc = wmma4_f32(a, b, acc);
    }
    const float bv = bias[colbase + l];
    const int row0 = rowbase + half * 8;
    #pragma unroll
    for (int j = 0; j < 8; ++j)
        out[(size_t)(row0 + j) * C_DIM + colbase + l] = acc[j] + bv;
}

// ---------------- 2) sim strip: sim[n,m] = q[n,:] . k[m,:] ----------------
// grid (N/512, STRIP/16), block 256 (8 waves; each wave owns FOUR 16-col tiles -> A reused 4x)
__global__ void sim_kernel(const float* __restrict__ q, const float* __restrict__ k,
                           float* __restrict__ sim, int strip_base) {
    __shared__ float qs[16 * C_DIM];             // 16 KB q strip, shared by 8 waves
    const int tid  = threadIdx.x;
    const int wave = tid >> 5;
    const int lane = tid & 31;
    const int half = lane >> 4;
    const int l    = lane & 15;
    const int nbase  = strip_base + blockIdx.y * 16;
    const int mbase0 = (blockIdx.x * 32 + wave * 4) * 16;   // first of 4 col tiles

    // Stage the q strip into LDS with CDNA5 async copies (ASYNCcnt-tracked,
    // DMA writes LDS directly — no VGPR round trip). 16 KB = 1024 x b128.
    {
        const char* gsrc = (const char*)(q + (size_t)nbase * C_DIM);
        #pragma unroll
        for (int i = 0; i < 4; ++i) {
            unsigned int      ldst = (unsigned int)(size_t)(qs) + (unsigned)(i * 256 + tid) * 16u;
            unsigned long long ga  = (unsigned long long)(size_t)(gsrc) + (unsigned long long)(i * 256 + tid) * 16ull;
            asm volatile("global_load_async_to_lds_b128 %0, %1, off"
                         :: "v"(ldst), "v"(ga) : "memory");
        }
        asm volatile("s_wait_asynccnt 0x0" ::: "memory");
    }
    __syncthreads();

    v8f acc0 = {}, acc1 = {}, acc2 = {}, acc3 = {};
    const float* k0 = k + (size_t)(mbase0 +  0 + l) * C_DIM;
    const float* k1 = k + (size_t)(mbase0 + 16 + l) * C_DIM;
    const float* k2 = k + (size_t)(mbase0 + 32 + l) * C_DIM;
    const float* k3 = k + (size_t)(mbase0 + 48 + l) * C_DIM;
    const float* qrow = qs + l * C_DIM;
    #pragma unroll 4
    for (int kk = 0; kk < C_DIM; kk += 4) {
        const int ko = kk + 2 * half;
        v2f a  = *(const v2f*)(qrow + ko);       // A: q tile (16xK), reused 4x
        v2f b0 = *(const v2f*)(k0 + ko);         // B: k^T tiles (Kx16)
        v2f b1 = *(const v2f*)(k1 + ko);
        v2f b2 = *(const v2f*)(k2 + ko);
        v2f b3 = *(const v2f*)(k3 + ko);
        acc0 = wmma4_f32(a, b0, acc0);
        acc1 = wmma4_f32(a, b1, acc1);
        acc2 = wmma4_f32(a, b2, acc2);
        acc3 = wmma4_f32(a, b3, acc3);
    }
    const int lrow0 = blockIdx.y * 16 + half * 8;   // row within strip
    #pragma unroll
    for (int j = 0; j < 8; ++j) {
        float* dst = sim + (size_t)(lrow0 + j) * N_PTS + mbase0 + l;
        dst[ 0] = acc0[j];
        dst[16] = acc1[j];
        dst[32] = acc2[j];
        dst[48] = acc3[j];
    }
}

// ---------------- 3) sorted top-128 per row (ties -> lowest index, like lax.top_k) ----------------
// grid STRIP, block 256
__global__ void topk128_kernel(const float* __restrict__ sim, int strip_base,
                               float* __restrict__ cscore, int* __restrict__ cidx) {
    __shared__ float vals[N_PTS];                // 32 KB row copy
    __shared__ float rv[256];
    __shared__ int   ri[256];
    const int t   = threadIdx.x;
    const int row = strip_base + blockIdx.x;
    const float* src = sim + (size_t)blockIdx.x * N_PTS;
    #pragma unroll 8
    for (int i = 0; i < 32; ++i) vals[i * 256 + t] = src[i * 256 + t];
    __syncthreads();

    for (int it = 0; it < KC; ++it) {
        float bv = -__builtin_inff();
        int   bi = 0x7fffffff;
        #pragma unroll 8
        for (int i = 0; i < 32; ++i) {
            const int j = i * 256 + t;
            const float x = vals[j];
            if (x > bv || (x == bv && j < bi)) { bv = x; bi = j; }
        }
        rv[t] = bv; ri[t] = bi;
        __syncthreads();
        for (int s = 128; s > 0; s >>= 1) {
            if (t < s) {
                const float xv = rv[t + s]; const int xi = ri[t + s];
                if (xv > rv[t] || (xv == rv[t] && xi < ri[t])) { rv[t] = xv; ri[t] = xi; }
            }
            __syncthreads();
        }
        if (t == 0) {
            cscore[(size_t)row * KC + it] = rv[0];
            cidx  [(size_t)row * KC + it] = ri[0];
            vals[ri[0]] = -__builtin_inff();
        }
        __syncthreads();
    }
}

// ---------------- 4) zero the 256 MB output (harness poisons it) ----------------
__global__ void zero_kernel(float4* __restrict__ P, size_t n4) {
    size_t i = (size_t)blockIdx.x * blockDim.x + threadIdx.x;
    const size_t stride = (size_t)gridDim.x * blockDim.x;
    const float4 z = {0.f, 0.f, 0.f, 0.f};
    for (; i < n4; i += stride) P[i] = z;
}

// ---------------- 5) mask + fine top-16 + softmax + scatter ----------------
// grid N, block 128 (one thread per coarse candidate)
__global__ void fine_kernel(const float* __restrict__ cscore, const int* __restrict__ cidx,
                            const float* __restrict__ dirs_a, const float* __restrict__ dirs_b,
                            float* __restrict__ P) {
    __shared__ int   s_m[KC];
    __shared__ float s_sc[KC];
    const int t   = threadIdx.x;
    const int row = blockIdx.x;

    const float dax = dirs_a[row * 3 + 0];
    const float day = dirs_a[row * 3 + 1];
    const float daz = dirs_a[row * 3 + 2];

    const int   idx = cidx  [(size_t)row * KC + t];
    const float sc  = cscore[(size_t)row * KC + t];
    const float bx = dirs_b[(size_t)idx * 3 + 0];
    const float by = dirs_b[(size_t)idx * 3 + 1];
    const float bz = dirs_b[(size_t)idx * 3 + 2];
    float cosv = dax * bx + day * by + daz * bz;
    cosv = fminf(1.0f, fmaxf(-1.0f, cosv));
    const int m = (acosf(cosv) <= LOCAL_RAD_F) ? 1 : 0;
    s_m[t] = m;
    __syncthreads();

    int any = 0;
    for (int i = 0; i < KC; ++i) any |= s_m[i];
    const int eff = any ? m : 1;                  // per-row fallback (covers global fallback too)
    int rank = 0;                                 // eff-true candidates before me (scores sorted desc)
    for (int i = 0; i < t; ++i) rank += any ? s_m[i] : 1;
    const int sel = eff && (rank < KF);           // fine top-16 == first <=16 eff-true in sorted order

    s_sc[t] = sel ? sc * TEMP_INV : -__builtin_inff();
    __syncthreads();

    float mx = -__builtin_inff();
    for (int i = 0; i < KC; ++i) mx = fmaxf(mx, s_sc[i]);
    float Z = 0.0f;
    for (int i = 0; i < KC; ++i) {
        const float v = s_sc[i];
        if (v != -__builtin_inff()) Z += expf(v - mx);
    }
    if (sel) {
        // masked-out fine slots hold NEG -> exp underflows to 0, so row sum == 1; ref divides by (1+1e-8)
        const float p = (expf(s_sc[t] - mx) / Z) / (1.0f + 1e-8f);
        P[(size_t)row * N_PTS + idx] = p;
    }
}

// ---------------- launch ----------------
extern "C" void kernel_launch(void* const* d_in, const int* in_sizes, int n_in,
                              void* d_out, int out_size, void* d_ws, size_t ws_size,
                              hipStream_t stream) {
    (void)in_sizes; (void)n_in; (void)out_size; (void)ws_size;
    const float* fa = (const float*)d_in[0];
    const float* fb = (const float*)d_in[1];
    const float* da = (const float*)d_in[2];
    const float* db = (const float*)d_in[3];
    const float* Wq = (const float*)d_in[4];
    const float* bq = (const float*)d_in[5];
    const float* Wk = (const float*)d_in[6];
    const float* bk = (const float*)d_in[7];
    float* P = (float*)d_out;

    char* ws = (char*)d_ws;                          // 56 MB total
    float* qb   = (float*)(ws);                      //  8 MB  q
    float* kb   = (float*)(ws + ((size_t)8  << 20)); //  8 MB  k
    float* simb = (float*)(ws + ((size_t)16 << 20)); // 32 MB  sim strip (1024 x 8192)
    float* csc  = (float*)(ws + ((size_t)48 << 20)); //  4 MB  coarse scores
    int*   cix  = (int*)  (ws + ((size_t)52 << 20)); //  4 MB  coarse indices

    proj_kernel<<<dim3(C_DIM / 16, N_PTS / 16, 2), 32, 0, stream>>>(fa, fb, Wq, bq, Wk, bk, qb, kb);

    for (int s = 0; s < N_PTS / STRIP; ++s) {
        const int base = s * STRIP;
        sim_kernel    <<<dim3(N_PTS / 512, STRIP / 16), 256, 0, stream>>>(qb, kb, simb, base);
        topk128_kernel<<<STRIP, 256, 0, stream>>>(simb, base, csc, cix);
    }

    zero_kernel<<<2048, 256, 0, stream>>>((float4*)P, (size_t)N_PTS * N_PTS / 4);
    fine_kernel<<<N_PTS, KC, 0, stream>>>(csc, cix, da, db, P);
}